// SparseSelfAttention_68676527063661
// MI455X (gfx1250) — compile-verified
//
#include <hip/hip_runtime.h>

#define BATCH  2
#define SEQ    4096
#define DMODEL 512
#define NHEADS 8
#define HDK    64
#define HDV    64
#define DRATE  4
#define NQ     1025   // phase length: (4096+4)/4
#define NP     1040   // padded to 65*16
#define QTILES 65
#define RTILES 260    // 4160 rows (= NP*DRATE) per batch for projection

#define WSZ ((size_t)BATCH * NHEADS * DRATE * NP * HDK)   // floats per tensor

typedef __attribute__((ext_vector_type(2))) float v2f;
typedef __attribute__((ext_vector_type(8))) float v8f;

__device__ __forceinline__ v8f wmma4(v2f a, v2f b, v8f c) {
  // D = A(16x4) * B(4x16) + C(16x16), all f32
  return __builtin_amdgcn_wmma_f32_16x16x4_f32(false, a, false, b, (short)0, c,
                                               false, false);
}

// ---------------------------------------------------------------------------
// Kernel 1: Q/K/V projection GEMM -> phase-major workspace [b][h][r][ni][64]
// One wave computes a 16x64 tile (one head slice) for one of Q/K/V.
// Padding tiles (t >= SEQ) are zero-filled (branch-free, block-uniform).
// ---------------------------------------------------------------------------
__global__ __launch_bounds__(32) void proj_kernel(
    const float* __restrict__ x, const float* __restrict__ Wq,
    const float* __restrict__ Wk, const float* __restrict__ Wv,
    float* __restrict__ ws) {
  const int lane = threadIdx.x;
  const int half = lane >> 4;
  const int l15  = lane & 15;

  int gid = blockIdx.x;
  const int h       = gid % NHEADS; gid /= NHEADS;
  const int w       = gid % 3;      gid /= 3;       // 0=Q 1=K 2=V
  const int rowtile = gid % RTILES; gid /= RTILES;
  const int b       = gid;

  const int t0 = rowtile * 16;
  float* dst = ws + (size_t)w * WSZ;

  if (t0 >= SEQ) {
    // zero padding rows so the attention kernel never reads poison
#pragma unroll
    for (int g = 0; g < 8; ++g) {
      const int t  = t0 + g + (half << 3);
      const int ni = t >> 2;
      const int rr = t & 3;
      const size_t base =
          (((size_t)(b * NHEADS + h) * DRATE + rr) * NP + ni) * HDK;
#pragma unroll
      for (int nc = 0; nc < 4; ++nc) dst[base + nc * 16 + l15] = 0.f;
    }
    return;
  }

  const float* W = (w == 0) ? Wq : (w == 1) ? Wk : Wv;
  const float* xrow = x + ((size_t)b * SEQ + t0 + l15) * DMODEL;
  const float* Wc   = W + h * HDK + l15;   // per-lane column base

  v8f acc[4];
#pragma unroll
  for (int nc = 0; nc < 4; ++nc)
    acc[nc] = (v8f){0.f, 0.f, 0.f, 0.f, 0.f, 0.f, 0.f, 0.f};

#pragma unroll 4
  for (int kk = 0; kk < DMODEL; kk += 4) {
    const int kd = kk + (half << 1);
    const v2f a = *(const v2f*)(xrow + kd);          // global_load_b64
    const float* w0 = Wc + (size_t)kd * DMODEL;
#pragma unroll
    for (int nc = 0; nc < 4; ++nc) {
      v2f bf;
      bf.x = w0[nc * 16];
      bf.y = w0[DMODEL + nc * 16];
      acc[nc] = wmma4(a, bf, acc[nc]);
    }
  }

#pragma unroll
  for (int g = 0; g < 8; ++g) {
    const int t  = t0 + g + (half << 3);
    const int ni = t >> 2;
    const int rr = t & 3;
    const size_t base =
        (((size_t)(b * NHEADS + h) * DRATE + rr) * NP + ni) * HDK;
#pragma unroll
    for (int nc = 0; nc < 4; ++nc) dst[base + nc * 16 + l15] = acc[nc][g];
  }
}

// ---------------------------------------------------------------------------
// Kernel 2: per-(b,h,r) flash attention over the phase + 7 patch logits.
// Softmax stats kept in C-layout-native form: m_i[g]/l_i[g] hold the stats
// for row (g + 8*half), uniform across each 16-lane group (no broadcasts).
// ---------------------------------------------------------------------------
__global__ __launch_bounds__(32) void attn_kernel(const float* __restrict__ ws,
                                                  float* __restrict__ out) {
  __shared__ float pbuf[16 * 16];
  const int lane = threadIdx.x;
  const int half = lane >> 4;
  const int l15  = lane & 15;
  const float scale = 0.125f;   // 1/sqrt(64)

  int gid = blockIdx.x;
  const int qt = gid % QTILES; gid /= QTILES;
  const int r  = gid % DRATE;  gid /= DRATE;
  const int h  = gid % NHEADS; gid /= NHEADS;
  const int b  = gid;

  const size_t phase = ((size_t)(b * NHEADS + h) * DRATE + r) * (size_t)NP * HDK;
  const float* Q = ws + phase;
  const float* K = ws + WSZ + phase;
  const float* V = ws + 2 * WSZ + phase;
  const int q0 = qt * 16;

  // resident Q fragments (A layout: lane=M, VGPR0 K in {0,2}, VGPR1 K in {1,3})
  v2f qf[16];
  {
    const float* qrow = Q + (size_t)(q0 + l15) * HDK;
#pragma unroll
    for (int c = 0; c < 16; ++c)
      qf[c] = *(const v2f*)(qrow + 4 * c + (half << 1));
  }

  v8f oacc[4];
#pragma unroll
  for (int nc = 0; nc < 4; ++nc)
    oacc[nc] = (v8f){0.f, 0.f, 0.f, 0.f, 0.f, 0.f, 0.f, 0.f};
  float m_i[8], l_i[8];
#pragma unroll
  for (int g = 0; g < 8; ++g) { m_i[g] = -1e30f; l_i[g] = 0.f; }

  for (int mt = 0; mt < QTILES; ++mt) {
    const int m0 = mt * 16;

    // ---- S = Q * K^T  (B: lane = key column, rows = head-dim) ----
    v8f s = {0.f, 0.f, 0.f, 0.f, 0.f, 0.f, 0.f, 0.f};
    const float* krow = K + (size_t)(m0 + l15) * HDK;
#pragma unroll
    for (int c = 0; c < 16; ++c) {
      const v2f bf = *(const v2f*)(krow + 4 * c + (half << 1));
      s = wmma4(qf[c], bf, s);
    }

    // scale; mask nonexistent keys only in the (uniform) last tile
    if (mt == QTILES - 1) {
      const bool cv = (m0 + l15) < NQ;
#pragma unroll
      for (int g = 0; g < 8; ++g) s[g] = cv ? s[g] * scale : -1e30f;
    } else {
#pragma unroll
      for (int g = 0; g < 8; ++g) s[g] *= scale;
    }

    // ---- online softmax, fully elementwise in C-layout ----
#pragma unroll
    for (int g = 0; g < 8; ++g) {
      float v = s[g];
      v = fmaxf(v, __shfl_xor(v, 1, 32));
      v = fmaxf(v, __shfl_xor(v, 2, 32));
      v = fmaxf(v, __shfl_xor(v, 4, 32));
      v = fmaxf(v, __shfl_xor(v, 8, 32));      // row max, uniform in group
      const float mn = fmaxf(m_i[g], v);
      const float al = __expf(m_i[g] - mn);
      const float p  = __expf(s[g] - mn);
      s[g] = p;
      float sum = p;
      sum += __shfl_xor(sum, 1, 32);
      sum += __shfl_xor(sum, 2, 32);
      sum += __shfl_xor(sum, 4, 32);
      sum += __shfl_xor(sum, 8, 32);           // row sum, uniform in group
      l_i[g] = l_i[g] * al + sum;
      m_i[g] = mn;
      oacc[0][g] *= al; oacc[1][g] *= al; oacc[2][g] *= al; oacc[3][g] *= al;
    }

    // ---- C-layout -> A-layout for P via LDS ----
    __syncthreads();
#pragma unroll
    for (int g = 0; g < 8; ++g)
      pbuf[(g + (half << 3)) * 16 + l15] = s[g];
    __syncthreads();
    v2f pa[4];
#pragma unroll
    for (int kc = 0; kc < 4; ++kc) {
      const int kd = 4 * kc + (half << 1);
      pa[kc].x = pbuf[l15 * 16 + kd];
      pa[kc].y = pbuf[l15 * 16 + kd + 1];
    }

    // ---- O += P * V  (unconditional: padding rows are zeroed) ----
#pragma unroll
    for (int nc = 0; nc < 4; ++nc) {
#pragma unroll
      for (int kc = 0; kc < 4; ++kc) {
        const int key = m0 + 4 * kc + (half << 1);
        v2f vb;
        vb.x = V[(size_t)key * HDK + nc * 16 + l15];
        vb.y = V[(size_t)(key + 1) * HDK + nc * 16 + l15];
        oacc[nc] = wmma4(pa[kc], vb, oacc[nc]);
      }
    }
  }

  // ---- 7 patch logits: same-phase neighbors ni-3..ni+3; OOB -> zero key ----
  const int qrow = q0 + l15;
  float plog[7];
#pragma unroll
  for (int j = 0; j < 7; ++j) {
    const int ni2  = qrow + j - 3;
    const bool inb = (unsigned)ni2 < (unsigned)NQ;
    const int nic  = inb ? ni2 : 0;              // clamped, load-safe
    const float* kr = K + (size_t)nic * HDK;
    float part = 0.f;
#pragma unroll
    for (int c = 0; c < 16; ++c) {
      const v2f kv = *(const v2f*)(kr + 4 * c + (half << 1));
      part += qf[c].x * kv.x + qf[c].y * kv.y;
    }
    const float dot = part + __shfl_xor(part, 16, 32);
    plog[j] = inb ? dot * scale : 0.f;           // zero key => logit exactly 0
  }

#pragma unroll
  for (int g = 0; g < 8; ++g) {
    const int srcl = half ? (g + 8) : g;
    float pl[7];
#pragma unroll
    for (int j = 0; j < 7; ++j) pl[j] = __shfl(plog[j], srcl, 32);
    float pm = pl[0];
#pragma unroll
    for (int j = 1; j < 7; ++j) pm = fmaxf(pm, pl[j]);
    const float mn = fmaxf(m_i[g], pm);
    const float al = __expf(m_i[g] - mn);
    float wj[7];
    float sum = 0.f;
#pragma unroll
    for (int j = 0; j < 7; ++j) { wj[j] = __expf(pl[j] - mn); sum += wj[j]; }
    l_i[g] = l_i[g] * al + sum;
    m_i[g] = mn;

    const int nie = q0 + g + (half << 3);
#pragma unroll
    for (int nc = 0; nc < 4; ++nc) {
      float o = oacc[nc][g] * al;
#pragma unroll
      for (int j = 0; j < 7; ++j) {
        const int ni2   = nie + j - 3;
        const bool inb2 = (unsigned)ni2 < (unsigned)NQ;
        const int nic2  = inb2 ? ni2 : 0;        // clamped, load-safe
        const float vv  = V[(size_t)nic2 * HDK + nc * 16 + l15];
        o += (inb2 ? wj[j] : 0.f) * vv;
      }
      oacc[nc][g] = o;
    }
  }

  // ---- normalize and write out (only t < SEQ) ----
#pragma unroll
  for (int g = 0; g < 8; ++g) {
    const int t = (q0 + g + (half << 3)) * DRATE + r;
    if (t < SEQ) {
      const float inv = 1.f / l_i[g];
      const size_t ob =
          ((size_t)b * SEQ + t) * (size_t)(NHEADS * HDV) + (size_t)h * HDV;
#pragma unroll
      for (int nc = 0; nc < 4; ++nc)
        out[ob + nc * 16 + l15] = oacc[nc][g] * inv;
    }
  }
}

// ---------------------------------------------------------------------------
extern "C" void kernel_launch(void* const* d_in, const int* in_sizes, int n_in,
                              void* d_out, int out_size, void* d_ws,
                              size_t ws_size, hipStream_t stream) {
  const float* x  = (const float*)d_in[0];
  const float* Wq = (const float*)d_in[1];
  const float* Wk = (const float*)d_in[2];
  const float* Wv = (const float*)d_in[3];
  float* ws  = (float*)d_ws;
  float* out = (float*)d_out;

  const int proj_blocks = BATCH * RTILES * 3 * NHEADS;     // 12480
  const int attn_blocks = BATCH * NHEADS * DRATE * QTILES; // 4160
  proj_kernel<<<proj_blocks, 32, 0, stream>>>(x, Wq, Wk, Wv, ws);
  attn_kernel<<<attn_blocks, 32, 0, stream>>>(ws, out);
}